// AGG_OP_47373489275373
// MI455X (gfx1250) — compile-verified
//
#include <hip/hip_runtime.h>
#include <stdint.h>

// ---------------------------------------------------------------------------
// GCN SpMM:  out[dst] += w_e * rsqrt(degO[src]) * rsqrt(degI[dst]) * feat[src]
// L2-resident gather/scatter problem (feat+out = 51MB << 192MB L2).
// CDNA5 paths: TENSOR_LOAD_TO_LDS (TDM) double-buffered edge stream with
// s_wait_tensorcnt, global_prefetch_b8 on upcoming feature rows.
// Fallback: global_load_async_to_lds_b32 (+ s_wait_asynccnt) if no TDM builtin.
// ---------------------------------------------------------------------------

#define TILE 256

#if __has_builtin(__builtin_amdgcn_tensor_load_to_lds)
#define USE_TDM 1
#else
#define USE_TDM 0
#endif

typedef unsigned int u32x4 __attribute__((ext_vector_type(4)));
typedef int          i32x4 __attribute__((ext_vector_type(4)));
typedef int          i32x8 __attribute__((ext_vector_type(8)));

// -------- fallback: per-lane async copy to LDS (proven to assemble) --------
__device__ __forceinline__ void async_ld_b32(uint32_t ldsOff, const void* g) {
  asm volatile("global_load_async_to_lds_b32 %0, %1, off"
               :: "v"(ldsOff), "v"((uint64_t)(uintptr_t)g)
               : "memory");
}
__device__ __forceinline__ void wait_async0() {
  asm volatile("s_wait_asynccnt 0x0" ::: "memory");
}

#if USE_TDM
// -------- TDM: 1-D contiguous strip of n_elems 4-byte elements -> LDS ------
// D# per CDNA5 ISA. OOB reads beyond tensor_dim0
// return ZERO, which self-masks the tail tile (w=0 -> c=0, s=d=0).
__device__ __forceinline__ void tdm_load_1d(uint32_t lds_addr, const void* gptr,
                                            uint32_t n_elems, uint32_t tile_elems) {
  const uint64_t ga = (uint64_t)(uintptr_t)gptr;
  u32x4 g0;
  g0[0] = 1u;                                         // count=1 (valid user D#)
  g0[1] = lds_addr;                                   // lds_addr [63:32]
  g0[2] = (uint32_t)ga;                               // global_addr[31:0]
  g0[3] = ((uint32_t)(ga >> 32) & 0x01FFFFFFu)        // global_addr[56:32]
          | 0x80000000u;                              // type=2 ("image")

  i32x8 g1;
  g1[0] = (int)0x00020000u;                           // data_size=2 (4 bytes)
  g1[1] = (int)((n_elems & 0xFFFFu) << 16);           // tensor_dim0[15:0]
  g1[2] = (int)((n_elems >> 16) | (1u << 16));        // tensor_dim0[31:16] | tensor_dim1=1
  g1[3] = (int)(tile_elems << 16);                    // tile_dim0
  g1[4] = 1;                                          // tile_dim1=1 (one row)
  g1[5] = (int)n_elems;                               // tensor_dim0_stride lo
  g1[6] = 0;                                          // stride hi | dim1_stride lo
  g1[7] = 0;

  i32x4 z4 = {0, 0, 0, 0};
#if defined(__clang_major__) && __clang_major__ >= 23
  i32x8 z8 = {0, 0, 0, 0, 0, 0, 0, 0};
  __builtin_amdgcn_tensor_load_to_lds(g0, g1, z4, z4, z8, 0);
#else
  __builtin_amdgcn_tensor_load_to_lds(g0, g1, z4, z4, 0);
#endif
}
#endif  // USE_TDM

__global__ void __launch_bounds__(256)
zero_kernel(float4* __restrict__ out4, long n4, uint32_t* __restrict__ deg, long nd) {
  long i  = (long)blockIdx.x * blockDim.x + threadIdx.x;
  long st = (long)gridDim.x * blockDim.x;
  for (long k = i; k < n4; k += st) out4[k] = make_float4(0.f, 0.f, 0.f, 0.f);
  for (long k = i; k < nd; k += st) deg[k] = 0u;
}

__global__ void __launch_bounds__(256)
deg_kernel(const int* __restrict__ src, const int* __restrict__ dst,
           uint32_t* __restrict__ deg_out, uint32_t* __restrict__ deg_in, int E) {
  int e = blockIdx.x * 256 + threadIdx.x;
  if (e < E) {
    atomicAdd(&deg_out[src[e]], 1u);
    atomicAdd(&deg_in [dst[e]], 1u);
  }
}

// in-place: u32 degree -> float rsqrt(max(deg,1))
__global__ void __launch_bounds__(256)
inv_kernel(uint32_t* __restrict__ degs, int n2) {
  int i = blockIdx.x * 256 + threadIdx.x;
  if (i < n2) {
    uint32_t d = degs[i];
    float f = (float)(d ? d : 1u);
    ((float*)degs)[i] = rsqrtf(f);
  }
}

__global__ void __launch_bounds__(256)
agg_kernel(const float* __restrict__ feat, const float* __restrict__ w,
           const int* __restrict__ src, const int* __restrict__ dst,
           const float* __restrict__ invo, const float* __restrict__ invi,
           float* __restrict__ out, int E) {
  __shared__ int   ls [2][TILE];   // src indices
  __shared__ int   ldd[2][TILE];   // dst indices
  __shared__ float lc [2][TILE];   // edge weight -> coefficient (in place)

  const int tid  = threadIdx.x;
  const int lane = tid & 31;
  const int wave = tid >> 5;
  const int numTiles = (E + TILE - 1) / TILE;

  auto kick = [&](int buf, int tile) {
#if USE_TDM
    // TDM ignores EXEC: gate on a SCALAR condition so only wave 0 of the
    // block issues the three DMA descriptors (s_cbranch, not EXEC masking).
    if (__builtin_amdgcn_readfirstlane(wave) == 0) {
      const int base = tile * TILE;
      const uint32_t rem = (uint32_t)(E - base);
      tdm_load_1d((uint32_t)(uintptr_t)&ls [buf][0], src + base, rem, TILE);
      tdm_load_1d((uint32_t)(uintptr_t)&ldd[buf][0], dst + base, rem, TILE);
      tdm_load_1d((uint32_t)(uintptr_t)&lc [buf][0], w   + base, rem, TILE);
    }
#else
    int e = tile * TILE + tid;
    if (e < E) {
      async_ld_b32((uint32_t)(uintptr_t)&ls [buf][tid], src + e);
      async_ld_b32((uint32_t)(uintptr_t)&ldd[buf][tid], dst + e);
      async_ld_b32((uint32_t)(uintptr_t)&lc [buf][tid], w   + e);
    }
#endif
  };

  int cur = 0;
  int t = blockIdx.x;
  if (t < numTiles) kick(0, t);

  for (; t < numTiles; t += gridDim.x) {
#if USE_TDM
    if (__builtin_amdgcn_readfirstlane(wave) == 0)
      __builtin_amdgcn_s_wait_tensorcnt(0);   // only wave 0 has TDM in flight
#else
    wait_async0();
#endif
    __syncthreads();      // tile data visible to all waves; prev tile done

    int nt = t + gridDim.x;
    if (nt < numTiles) kick(cur ^ 1, nt);     // overlap next tile's DMA

    // --- coefficient phase: c_e = w_e * invsqrt(degO[s]) * invsqrt(degI[d])
    {
      int e = t * TILE + tid;
      int s = 0, d = 0; float c = 0.0f;
      if (e < E) {
        s = ls [cur][tid];
        d = ldd[cur][tid];
        float we = lc[cur][tid];
        c = we * invo[s] * invi[d];   // two random 4B L2 hits
      }
      ls [cur][tid] = s;
      ldd[cur][tid] = d;
      lc [cur][tid] = c;
    }
    __syncthreads();

    // --- scatter phase: each wave owns 32 edges; 32 lanes cover one 256B row
    const int base = wave * 32;
    for (int i = 0; i < 32; ++i) {
      int   s = ls [cur][base + i];   // uniform LDS broadcast reads
      int   d = ldd[cur][base + i];
      float c = lc [cur][base + i];
      if (i + 8 < 32) {               // prefetch a future source row into L2/L0
        int sp = ls[cur][base + i + 8];
        __builtin_prefetch(feat + (size_t)sp * 64 + 2 * lane, 0, 0);
      }
      const float2 f = *((const float2*)(feat + (size_t)s * 64) + lane);
      float* o = out + (size_t)d * 64 + 2 * lane;
      atomicAdd(o + 0, f.x * c);
      atomicAdd(o + 1, f.y * c);
    }
    // next iteration's top barrier protects buffer reuse
    cur ^= 1;
  }
}

extern "C" void kernel_launch(void* const* d_in, const int* in_sizes, int n_in,
                              void* d_out, int out_size, void* d_ws, size_t ws_size,
                              hipStream_t stream) {
  const float* feat = (const float*)d_in[0];
  const float* wgt  = (const float*)d_in[1];
  const int*   src  = (const int*)d_in[2];
  const int*   dst  = (const int*)d_in[3];
  float* out = (float*)d_out;

  const int N = in_sizes[0] / 64;   // feat is [N, 64]
  const int E = in_sizes[1];

  uint32_t* deg = (uint32_t*)d_ws;  // [0,N): deg_out, [N,2N): deg_in (->inv)

  zero_kernel<<<1024, 256, 0, stream>>>((float4*)out, (long)N * 16, deg, (long)2 * N);
  deg_kernel<<<(E + 255) / 256, 256, 0, stream>>>(src, dst, deg, deg + N, E);
  inv_kernel<<<(2 * N + 255) / 256, 256, 0, stream>>>(deg, 2 * N);

  const float* invo = (const float*)deg;
  const float* invi = invo + N;
  agg_kernel<<<1024, 256, 0, stream>>>(feat, wgt, src, dst, invo, invi, out, E);

  (void)n_in; (void)out_size; (void)ws_size;
}